// SymLoss_35338990911546
// MI455X (gfx1250) — compile-verified
//
#include <hip/hip_runtime.h>

typedef __attribute__((ext_vector_type(2))) float v2f;
typedef __attribute__((ext_vector_type(8))) float v8f;

#define RES      32
#define NPTS     65536
#define NB       64
#define REG_COEF 25.0f

// ---------------------------------------------------------------------------
// Kernel 1: regularizer term. One block of 64 threads (one per batch).
// Writes (initializes) d_out[0] = REG_COEF * sum_b ||N N^T - I||_F.
// ---------------------------------------------------------------------------
__global__ void __launch_bounds__(64) sym_reg_kernel(
    const float* __restrict__ planes, float* __restrict__ out) {
  __shared__ float red[NB];
  const int b = threadIdx.x;
  const float* p = planes + b * 3 * 4;

  float nh[3][3];
#pragma unroll
  for (int h = 0; h < 3; ++h) {
    float x = p[h * 4 + 0], y = p[h * 4 + 1], z = p[h * 4 + 2];
    float inv = 1.0f / sqrtf(x * x + y * y + z * z);
    nh[h][0] = x * inv; nh[h][1] = y * inv; nh[h][2] = z * inv;
  }
  float s = 0.0f;
#pragma unroll
  for (int i = 0; i < 3; ++i)
#pragma unroll
    for (int j = 0; j < 3; ++j) {
      float a = nh[i][0] * nh[j][0] + nh[i][1] * nh[j][1] + nh[i][2] * nh[j][2]
                - (i == j ? 1.0f : 0.0f);
      s += a * a;
    }
  red[b] = sqrtf(s);
  __syncthreads();
  if (b == 0) {
    float t = 0.0f;
#pragma unroll
    for (int i = 0; i < NB; ++i) t += red[i];
    out[0] = REG_COEF * t;   // initializes the poisoned output
  }
}

// ---------------------------------------------------------------------------
// Kernel 2: main reflection loss. grid = (32 chunks, 64 batches), 256 thr.
// Each wave handles 256 contiguous points (8 groups of 32). dist computed
// with v_wmma_f32_16x16x4_f32: A = planes (rows (nx,ny,nz,d), rows 3..15 = 0),
// B = 16 point columns (x,y,z,1). D tile: lane p, VGPR h holds dist[h][p]
// for lanes 0..15 -> owning-lane distribution; second tile shuffled up.
// ---------------------------------------------------------------------------
__global__ void __launch_bounds__(256) sym_main_kernel(
    const float* __restrict__ planes,
    const float* __restrict__ sp,
    const float* __restrict__ cp,
    float* __restrict__ out) {
  const int b    = blockIdx.y;
  const int tid  = threadIdx.x;
  const int lane = tid & 31;
  const int wave = tid >> 5;
  const bool lo  = lane < 16;
  const int r    = lane & 15;

  // --- per-batch plane constants (uniform across block, kept in regs) ---
  const float* pl = planes + b * 12;
  float nh[3][3], dpl[3];
#pragma unroll
  for (int h = 0; h < 3; ++h) {
    float x = pl[h * 4 + 0], y = pl[h * 4 + 1], z = pl[h * 4 + 2];
    float inv = 1.0f / sqrtf(x * x + y * y + z * z);
    nh[h][0] = x * inv; nh[h][1] = y * inv; nh[h][2] = z * inv;
    dpl[h] = pl[h * 4 + 3];
  }

  // --- A matrix (16x4 f32): lanes 0-15 hold K=0,1; lanes 16-31 hold K=2,3 ---
  v2f A;
  float a0_lo = (r == 0) ? nh[0][0] : (r == 1) ? nh[1][0] : (r == 2) ? nh[2][0] : 0.0f;
  float a1_lo = (r == 0) ? nh[0][1] : (r == 1) ? nh[1][1] : (r == 2) ? nh[2][1] : 0.0f;
  float a0_hi = (r == 0) ? nh[0][2] : (r == 1) ? nh[1][2] : (r == 2) ? nh[2][2] : 0.0f;
  float a1_hi = (r == 0) ? dpl[0]   : (r == 1) ? dpl[1]   : (r == 2) ? dpl[2]   : 0.0f;
  A.x = lo ? a0_lo : a0_hi;
  A.y = lo ? a1_lo : a1_hi;

  const float* spb = sp + (size_t)b * NPTS * 3;
  const float* cpb = cp + (size_t)b * RES * RES * RES * 3;
  const int base = blockIdx.x * 2048 + wave * 256;

  float acc = 0.0f;

#pragma unroll 1
  for (int i = 0; i < 8; ++i) {
    const int idx = base + i * 32 + lane;
    const float* ptp = spb + (size_t)idx * 3;
    // stream points non-temporally: keep L2 capacity for cp_grids gathers
    float px = __builtin_nontemporal_load(ptp + 0);
    float py = __builtin_nontemporal_load(ptp + 1);
    float pz = __builtin_nontemporal_load(ptp + 2);
    if (i < 7) __builtin_prefetch(ptp + 96, 0, 1);   // next 32-point group

    // --- B tiles (4x16 f32): lanes 0-15 = K0,K1 ; lanes 16-31 = K2,K3 ---
    float t_zlo = __shfl(pz, r, 32);        // z of point owned by lane-16
    float t_xhi = __shfl(px, r + 16, 32);   // x of point owned by lane+16
    float t_yhi = __shfl(py, r + 16, 32);   // y of point owned by lane+16
    v2f B1, B2;
    B1.x = lo ? px    : t_zlo;  B1.y = lo ? py    : 1.0f;
    B2.x = lo ? t_xhi : pz;     B2.y = lo ? t_yhi : 1.0f;

    v8f c0 = {};
    v8f D1 = __builtin_amdgcn_wmma_f32_16x16x4_f32(
        false, A, false, B1, (short)0, c0, false, false);
    v8f D2 = __builtin_amdgcn_wmma_f32_16x16x4_f32(
        false, A, false, B2, (short)0, c0, false, false);

#pragma unroll
    for (int h = 0; h < 3; ++h) {
      float d2s  = __shfl(D2[h], r, 32);           // move tile-2 dists up
      float dist = lo ? D1[h] : d2s;               // dist = p·n̂ + d
      float rx = px - 2.0f * dist * nh[h][0];
      float ry = py - 2.0f * dist * nh[h][1];
      float rz = pz - 2.0f * dist * nh[h][2];
      int vx = (int)fminf(fmaxf(floorf(rx * (float)RES), 0.0f), (float)(RES - 1));
      int vy = (int)fminf(fmaxf(floorf(ry * (float)RES), 0.0f), (float)(RES - 1));
      int vz = (int)fminf(fmaxf(floorf(rz * (float)RES), 0.0f), (float)(RES - 1));
      const float* g = cpb + ((size_t)((vx * RES + vy) * RES + vz)) * 3;
      float dx = rx - g[0];
      float dy = ry - g[1];
      float dz = rz - g[2];
      acc += sqrtf(dx * dx + dy * dy + dz * dz);
    }
  }

  // --- block reduction + one atomic per block ---
  __shared__ float red[256];
  red[tid] = acc;
  __syncthreads();
#pragma unroll
  for (int s = 128; s > 0; s >>= 1) {
    if (tid < s) red[tid] += red[tid + s];
    __syncthreads();
  }
  if (tid == 0) atomicAdd(out, red[0] * (1.0f / (float)NPTS));
}

// ---------------------------------------------------------------------------
extern "C" void kernel_launch(void* const* d_in, const int* in_sizes, int n_in,
                              void* d_out, int out_size, void* d_ws, size_t ws_size,
                              hipStream_t stream) {
  const float* planes = (const float*)d_in[0];   // (64, 3, 4)
  const float* sp     = (const float*)d_in[1];   // (64, 65536, 3)
  // d_in[2] = voxel_grids: unused by the reference math
  const float* cpg    = (const float*)d_in[3];   // (64, 32, 32, 32, 3)
  float* out = (float*)d_out;

  // 1) init d_out with the regularizer term
  sym_reg_kernel<<<1, 64, 0, stream>>>(planes, out);
  // 2) accumulate reflection loss: 32 chunks x 64 batches, 256 threads
  dim3 grid(32, NB);
  sym_main_kernel<<<grid, 256, 0, stream>>>(planes, sp, cpg, out);
}